// PCEN_54949811585300
// MI455X (gfx1250) — compile-verified
//
#include <hip/hip_runtime.h>
#include <math.h>

// PCEN for MI455X (gfx1250): one wave32 per (b,f) row of length T=4096.
// filtfilt EMA done as scan-by-matmul: 16x16 lower-triangular decay matrix
// times 16x256 reshaped row via V_WMMA_F32_16X16X4_F32 (f32 WMMA, exact),
// inter-chunk carries via wave32 affine shuffle-scan, carry correction fused
// into the WMMA epilogue.

typedef float v2f __attribute__((ext_vector_type(2)));
typedef float v8f __attribute__((ext_vector_type(8)));

#define T_LEN   4096
#define F_DIM   256
#define NCH     256      // chunks of 16 along T
#define CH_PAD  17       // padded chunk stride (bank-conflict-free)
#define LDSF    (NCH * CH_PAD)   // 4352 floats per row buffer

#define EPS_INV 1.0e6f
#define LOG_EPS (-13.8155105579642741f)   // ln(1e-6)

__device__ __forceinline__ int padAddr(int t) { return (t >> 4) * CH_PAD + (t & 15); }

// virtual index u -> physical padded LDS address (rev reads the row reversed)
__device__ __forceinline__ int vAddr(int u, bool rev) {
    int t = rev ? (T_LEN - 1 - u) : u;
    return (t >> 4) * CH_PAD + (t & 15);
}

__global__ __launch_bounds__(32) void pcen_kernel(
    const float* __restrict__ x,
    const float* __restrict__ log_alpha,
    const float* __restrict__ log_delta,
    const float* __restrict__ log_r,
    const float* __restrict__ z_ks,
    float* __restrict__ out)
{
    __shared__ float sx[LDSF];   // input row (forward order, padded)
    __shared__ float sy[LDSF];   // forward-smoothed row per scale
    __shared__ float sm[LDSF];   // accumulated M = sum_k w_k * smoother_k
    __shared__ float swb[NCH];   // per-chunk local tail w[c]
    __shared__ float sfb[NCH];   // per-chunk carry-in f[c]

    const int  lane = threadIdx.x;
    const int  row  = blockIdx.x;           // b*F + f   (C == 1)
    const int  f    = row & (F_DIM - 1);
    const long base = (long)row * T_LEN;

    // ---- stage x row into padded LDS; zero M accumulator -------------------
    __builtin_prefetch(x + base + lane * 128, 0, 1);   // global_prefetch_b8
    #pragma unroll 4
    for (int q = 0; q < T_LEN / (32 * 4); ++q) {
        const int t4 = (q * 32 + lane) * 4;
        const float4 v = *(const float4*)(x + base + t4);
        const int a = padAddr(t4);           // 4 consecutive (never crosses chunk)
        sx[a + 0] = v.x; sx[a + 1] = v.y; sx[a + 2] = v.z; sx[a + 3] = v.w;
    }
    for (int q = lane; q < LDSF; q += 32) sm[q] = 0.f;

    // ---- softmax denominator over ALL K*F entries of z_ks ------------------
    float tot = 0.f;
    for (int idx = lane; idx < 4 * F_DIM; idx += 32) tot += __expf(z_ks[idx]);
    #pragma unroll
    for (int d = 16; d >= 1; d >>= 1) tot += __shfl_xor(tot, d, 32);
    const float invTot = 1.0f / tot;

    const float alpha = __expf(log_alpha[f]);
    const float delta = __expf(log_delta[f]);
    const float rr    = __expf(log_r[f]);

    const int  Ln   = lane & 15;
    const bool loH  = lane < 16;
    const int  off  = loH ? 0 : 2;     // A/B K-halves per ISA WMMA layout
    const int  off8 = loH ? 0 : 8;     // C/D M-halves per ISA WMMA layout

    const float sList[4] = {0.015f, 0.02f, 0.04f, 0.08f};

    __syncthreads();

    for (int k = 0; k < 4; ++k) {
        const float s   = sList[k];
        const float a   = 1.0f - s;
        const float lna = __logf(a);
        const float wk  = __expf(z_ks[k * F_DIM + f]) * invTot;

        // A slices of T[i][j] = (j<=i) ? s*a^(i-j) : 0 ; lane holds row Ln,
        // K pair (j0+off, j0+off+1)  -- shared by fwd and bwd passes.
        v2f A0, A1, A2, A3;
        {
            const int i = Ln;
            #define TF(jj) (((jj) <= i) ? s * __expf(lna * (float)(i - (jj))) : 0.f)
            A0.x = TF(0  + off); A0.y = TF(1  + off);
            A1.x = TF(4  + off); A1.y = TF(5  + off);
            A2.x = TF(8  + off); A2.y = TF(9  + off);
            A3.x = TF(12 + off); A3.y = TF(13 + off);
            #undef TF
        }
        const float a16   = __expf(16.f  * lna);
        const float a16p8 = __expf(128.f * lna);
        const float corr0 = __expf(lna * (float)(off8 + 1));  // a^(off8+1)

        // two passes: pass 0 = forward (sx -> sy); pass 1 = backward
        // (sy read reversed -> accumulate wk * z into sm)
        for (int pass = 0; pass < 2; ++pass) {
            const bool rev = (pass == 1);
            const float* in = rev ? sy : sx;

            // -- Phase 1: chunk tails w[c] = s * sum_j a^(15-j) * in[c*16+j]
            for (int c = lane; c < NCH; c += 32) {
                float h = 0.f;
                const int ub = c * 16;
                #pragma unroll
                for (int j = 0; j < 16; ++j) h = h * a + in[vAddr(ub + j, rev)];
                swb[c] = s * h;
            }
            __syncthreads();

            // -- Phase 2: carries f[0]=boundary elem; f[c]=a16*f[c-1]+w[c-1]
            {
                const int cb = lane * 8;
                float W = 0.f;
                #pragma unroll
                for (int q = 0; q < 8; ++q) W = W * a16 + swb[cb + q];
                float Ai = a16p8, Bi = W;           // segment transform
                #pragma unroll
                for (int d = 1; d < 32; d <<= 1) {  // inclusive affine scan
                    const float pA = __shfl_up(Ai, d, 32);
                    const float pB = __shfl_up(Bi, d, 32);
                    if (lane >= d) { Bi = Ai * pB + Bi; Ai = Ai * pA; }
                }
                float eA = __shfl_up(Ai, 1, 32);    // exclusive
                float eB = __shfl_up(Bi, 1, 32);
                if (lane == 0) { eA = 1.f; eB = 0.f; }
                const float f0 = in[vAddr(0, rev)]; // lfilter_zi: carry = x[0]
                float fc = eA * f0 + eB;
                #pragma unroll
                for (int q = 0; q < 8; ++q) { sfb[cb + q] = fc; fc = a16 * fc + swb[cb + q]; }
            }
            __syncthreads();

            // -- Phase 3: WMMA local matmul + fused carry correction + output
            for (int tile = 0; tile < 16; ++tile) {
                const int cB = tile * 16 + Ln;
                v8f acc = {0.f, 0.f, 0.f, 0.f, 0.f, 0.f, 0.f, 0.f};
                #pragma unroll
                for (int k4 = 0; k4 < 4; ++k4) {
                    const int u0 = cB * 16 + 4 * k4 + off;
                    v2f B; B.x = in[vAddr(u0, rev)]; B.y = in[vAddr(u0 + 1, rev)];
                    const v2f As = (k4 == 0) ? A0 : (k4 == 1) ? A1 : (k4 == 2) ? A2 : A3;
                    acc = __builtin_amdgcn_wmma_f32_16x16x4_f32(
                        false, As, false, B, (short)0, acc, false, false);
                }
                const float fcv = sfb[cB];
                float coef = corr0;                  // a^(i+1), i = m + off8
                if (!rev) {
                    const int wa = cB * CH_PAD + off8;
                    #pragma unroll
                    for (int m = 0; m < 8; ++m) { sy[wa + m] = acc[m] + coef * fcv; coef *= a; }
                } else {
                    // z at virtual u maps to physical t = 4095-u
                    const int wa = (255 - cB) * CH_PAD + (15 - off8);
                    #pragma unroll
                    for (int m = 0; m < 8; ++m) { sm[wa - m] += wk * (acc[m] + coef * fcv); coef *= a; }
                }
            }
            __syncthreads();
        }
    }

    // ---- pointwise PCEN epilogue, coalesced float4 store -------------------
    const float dr = __powf(delta, rr);
    #pragma unroll 4
    for (int q = 0; q < T_LEN / (32 * 4); ++q) {
        const int t4 = (q * 32 + lane) * 4;
        const int a0 = padAddr(t4);
        float4 o;
        {
            const float g = __expf(-alpha * (LOG_EPS + log1pf(sm[a0 + 0] * EPS_INV)));
            o.x = __powf(sx[a0 + 0] * g + delta, rr) - dr;
        }
        {
            const float g = __expf(-alpha * (LOG_EPS + log1pf(sm[a0 + 1] * EPS_INV)));
            o.y = __powf(sx[a0 + 1] * g + delta, rr) - dr;
        }
        {
            const float g = __expf(-alpha * (LOG_EPS + log1pf(sm[a0 + 2] * EPS_INV)));
            o.z = __powf(sx[a0 + 2] * g + delta, rr) - dr;
        }
        {
            const float g = __expf(-alpha * (LOG_EPS + log1pf(sm[a0 + 3] * EPS_INV)));
            o.w = __powf(sx[a0 + 3] * g + delta, rr) - dr;
        }
        *(float4*)(out + base + t4) = o;
    }
}

extern "C" void kernel_launch(void* const* d_in, const int* in_sizes, int n_in,
                              void* d_out, int out_size, void* d_ws, size_t ws_size,
                              hipStream_t stream) {
    const float* x  = (const float*)d_in[0];
    const float* la = (const float*)d_in[1];
    const float* ld = (const float*)d_in[2];
    const float* lr = (const float*)d_in[3];
    const float* zk = (const float*)d_in[4];
    float* out = (float*)d_out;

    const int rows = 16 * 1 * F_DIM;   // B * C * F = 4096
    hipLaunchKernelGGL(pcen_kernel, dim3(rows), dim3(32), 0, stream,
                       x, la, ld, lr, zk, out);
}